// GodSlayer2_23167053595413
// MI455X (gfx1250) — compile-verified
//
#include <hip/hip_runtime.h>

typedef __attribute__((ext_vector_type(2))) float v2f;
typedef __attribute__((ext_vector_type(8))) float v8f;

#define LEN 256
#define PITCH 264      // 256 cols + 8-col periodic halo (4 each side)
#define NLAYERS 16

// up-dist 4..1, center, down-dist 1..4
#define ROWCALC(Q, A4, A3, A2, A1, CC, B1, B2, B3, B4, NC) {            \
    const float h1_ = Q[p-1] + Q[p+1], h2_ = Q[p-2] + Q[p+2];           \
    const float h3_ = Q[p-3] + Q[p+3], h4_ = Q[p-4] + Q[p+4];           \
    const float a_  = CC*w0 + w1*(h1_+A1+B1) + w2*(h2_+A2+B2)           \
                    + w3*(h3_+A3+B3) + w4*(h4_+A4+B4);                  \
    NC = CC + fmaxf(a_, 0.f) + expm1f(fminf(a_, 0.f)); }

// One workgroup per (batch, channel): 512 WGs x 256 threads (8 wave32).
// Whole 256x256 fp32 grid lives in LDS (pitch 264) for all 16 layers.
__launch_bounds__(256)
__global__ void stencil_evolve(const float* __restrict__ nsrc,
                               const float* __restrict__ weights,
                               float* __restrict__ chsum)
{
    __shared__ float xs[LEN][PITCH];   // ~264 KB state
    __shared__ float saved[4][LEN];    // old rows 0..3 (wrap) + reduce buffer

    const int tid  = threadIdx.x;      // column j (0..255)
    const int p    = tid + 4;          // padded column index
    const int lane = tid & 31;
    const int wid  = tid >> 5;         // wave 0..7
    const int b    = blockIdx.x >> 3;
    const int c    = blockIdx.x & 7;

    // ---- Gram g = n_b * n_b^T via V_WMMA_F32_16X16X4_F32 (K: 3 padded to 4) ----
    {
        const float* nb = nsrc + b * (LEN * 3);
        const int rsel = lane & 15;
        const int hi   = lane >> 4;    // 0 => holds K=0,1 ; 1 => K=2,pad

        v2f frag[16];                  // all 16 tile fragments, branchless
        #pragma unroll
        for (int t = 0; t < 16; ++t) {
            const int r  = t * 16 + rsel;
            const float vx = nb[3 * r + (hi << 1)];
            const float vy = nb[3 * r + 1];
            frag[t].x = vx;
            frag[t].y = hi ? 0.0f : vy;
        }
        #pragma unroll
        for (int mi = 0; mi < 2; ++mi) {
            const int mt = wid * 2 + mi;
            const int ra = mt * 16 + rsel;
            const float ax = nb[3 * ra + (hi << 1)];
            const float ay = nb[3 * ra + 1];
            v2f a;
            a.x = ax;
            a.y = hi ? 0.0f : ay;
            #pragma unroll
            for (int nt = 0; nt < 16; ++nt) {
                v8f acc = {0.f, 0.f, 0.f, 0.f, 0.f, 0.f, 0.f, 0.f};
                acc = __builtin_amdgcn_wmma_f32_16x16x4_f32(
                    false, a, false, frag[nt], (short)0, acc, false, false);
                #pragma unroll
                for (int v = 0; v < 8; ++v)
                    xs[mt * 16 + v + 8 * hi][nt * 16 + rsel + 4] = acc[v];
            }
            // Keep only one M-tile's accumulators in flight: caps the wave
            // below 256 VGPRs so no s_set_vgpr_msb addressing is needed.
            __builtin_amdgcn_sched_barrier(0);
        }
    }
    __syncthreads();
    xs[tid][p] = 0.0f;                 // zero diagonal (row tid, col tid)
    // Fill row tid's periodic halo (same-thread ordered with its diag write).
    xs[tid][0]   = xs[tid][256]; xs[tid][1]   = xs[tid][257];
    xs[tid][2]   = xs[tid][258]; xs[tid][3]   = xs[tid][259];
    xs[tid][260] = xs[tid][4];   xs[tid][261] = xs[tid][5];
    xs[tid][262] = xs[tid][6];   xs[tid][263] = xs[tid][7];

    float colsum = 0.0f;
    for (int layer = 0; layer < NLAYERS; ++layer) {
        __syncthreads();   // previous layer (and init) fully committed
        colsum = 0.0f;     // only the final layer's sum survives

        const float* wp = weights + ((layer << 3) + c) * 5;   // uniform -> SMEM
        const float w0 = wp[0], w1 = wp[1], w2 = wp[2], w3 = wp[3], w4 = wp[4];
        const bool commit = (layer != NLAYERS - 1);           // layer-15 stores are dead

        // Stash old rows 0..3 for the periodic wrap (column-private).
        saved[0][tid] = xs[0][p]; saved[1][tid] = xs[1][p];
        saved[2][tid] = xs[2][p]; saved[3][tid] = xs[3][p];

        // Register window: old rows i-4 .. i+11 at this column (i = 0).
        float r0  = xs[252][p], r1  = xs[253][p], r2  = xs[254][p], r3  = xs[255][p];
        float r4  = xs[0][p],   r5  = xs[1][p],   r6  = xs[2][p],   r7  = xs[3][p];
        float r8  = xs[4][p],   r9  = xs[5][p],   r10 = xs[6][p],   r11 = xs[7][p];
        float r12 = xs[8][p],   r13 = xs[9][p],   r14 = xs[10][p],  r15 = xs[11][p];

        // 8 rows per barrier.
        for (int i = 0; i < LEN; i += 8) {
            const float *q0 = xs[i],   *q1 = xs[i+1], *q2 = xs[i+2], *q3 = xs[i+3];
            const float *q4 = xs[i+4], *q5 = xs[i+5], *q6 = xs[i+6], *q7 = xs[i+7];
            float nc0, nc1, nc2, nc3, nc4, nc5, nc6, nc7;
            ROWCALC(q0, r0, r1, r2,  r3,  r4,  r5,  r6,  r7,  r8,  nc0);
            ROWCALC(q1, r1, r2, r3,  r4,  r5,  r6,  r7,  r8,  r9,  nc1);
            ROWCALC(q2, r2, r3, r4,  r5,  r6,  r7,  r8,  r9,  r10, nc2);
            ROWCALC(q3, r3, r4, r5,  r6,  r7,  r8,  r9,  r10, r11, nc3);
            ROWCALC(q4, r4, r5, r6,  r7,  r8,  r9,  r10, r11, r12, nc4);
            ROWCALC(q5, r5, r6, r7,  r8,  r9,  r10, r11, r12, r13, nc5);
            ROWCALC(q6, r6, r7, r8,  r9,  r10, r11, r12, r13, r14, nc6);
            ROWCALC(q7, r7, r8, r9,  r10, r11, r12, r13, r14, r15, nc7);
            colsum += ((nc0 + nc1) + (nc2 + nc3)) + ((nc4 + nc5) + (nc6 + nc7));

            // Slide-in old rows i+12..i+19 (untouched xs, or the stash on wrap).
            float s0, s1, s2, s3, s4, s5, s6, s7;
            if (i <= 232) {
                s0 = xs[i+12][p]; s1 = xs[i+13][p]; s2 = xs[i+14][p]; s3 = xs[i+15][p];
                s4 = xs[i+16][p]; s5 = xs[i+17][p]; s6 = xs[i+18][p]; s7 = xs[i+19][p];
            } else if (i == 240) {
                s0 = xs[252][p];   s1 = xs[253][p];   s2 = xs[254][p];   s3 = xs[255][p];
                s4 = saved[0][tid]; s5 = saved[1][tid]; s6 = saved[2][tid]; s7 = saved[3][tid];
            } else {
                s0 = s1 = s2 = s3 = s4 = s5 = s6 = s7 = 0.f;
            }

            __syncthreads();          // all reads of old rows i..i+7 complete

            if (commit) {             // uniform branch
                xs[i  ][p] = nc0; xs[i+1][p] = nc1; xs[i+2][p] = nc2; xs[i+3][p] = nc3;
                xs[i+4][p] = nc4; xs[i+5][p] = nc5; xs[i+6][p] = nc6; xs[i+7][p] = nc7;
                if (tid < 4) {        // mirror into right halo
                    xs[i  ][tid+260] = nc0; xs[i+1][tid+260] = nc1;
                    xs[i+2][tid+260] = nc2; xs[i+3][tid+260] = nc3;
                    xs[i+4][tid+260] = nc4; xs[i+5][tid+260] = nc5;
                    xs[i+6][tid+260] = nc6; xs[i+7][tid+260] = nc7;
                }
                if (tid >= 252) {     // mirror into left halo
                    xs[i  ][tid-252] = nc0; xs[i+1][tid-252] = nc1;
                    xs[i+2][tid-252] = nc2; xs[i+3][tid-252] = nc3;
                    xs[i+4][tid-252] = nc4; xs[i+5][tid-252] = nc5;
                    xs[i+6][tid-252] = nc6; xs[i+7][tid-252] = nc7;
                }
            }
            r0  = r8;  r1  = r9;  r2  = r10; r3  = r11;
            r4  = r12; r5  = r13; r6  = r14; r7  = r15;
            r8  = s0;  r9  = s1;  r10 = s2;  r11 = s3;
            r12 = s4;  r13 = s5;  r14 = s6;  r15 = s7;
        }
    }

    // colsum now holds the final layer's column sum (column-private saved[] reuse).
    saved[0][tid] = colsum;
    __syncthreads();
    for (int off = 128; off > 0; off >>= 1) {
        if (tid < off) saved[0][tid] += saved[0][tid + off];
        __syncthreads();
    }
    if (tid == 0) chsum[blockIdx.x] = saved[0][0];   // index = b*8 + c
}

__global__ void finalize(const float* __restrict__ chsum, float* __restrict__ out)
{
    const int t = threadIdx.x;
    if (t < 64) {
        float s = 0.0f;
        #pragma unroll
        for (int c = 0; c < 8; ++c) s += chsum[t * 8 + c];
        out[t] = expf(-s * (1.0f / 524288.0f));      // mean over 256*256*8
    }
}

extern "C" void kernel_launch(void* const* d_in, const int* in_sizes, int n_in,
                              void* d_out, int out_size, void* d_ws, size_t ws_size,
                              hipStream_t stream)
{
    const float* n = (const float*)d_in[0];   // (64, 256, 3) fp32
    const float* w = (const float*)d_in[1];   // (16, 8, 5) fp32
    float* ws  = (float*)d_ws;                // 512 partial sums (all overwritten)
    float* out = (float*)d_out;               // (64,) fp32

    stencil_evolve<<<dim3(64 * 8), dim3(256), 0, stream>>>(n, w, ws);
    finalize<<<dim3(1), dim3(64), 0, stream>>>(ws, out);
}